// ComplexGCN_15693810500042
// MI455X (gfx1250) — compile-verified
//
#include <hip/hip_runtime.h>
#include <stdint.h>

#define NN 50000
#define NE 800000
#define DD 128
#define TT 20

typedef __attribute__((ext_vector_type(2))) float v2f;
typedef __attribute__((ext_vector_type(8))) float v8f;

// Non-returning HW f32 atomic add at device scope (L2 atomic units).
// Tracked by STOREcnt; S_ENDPGM's implicit wait-idle guarantees completion
// before the next kernel in the stream reads the buffer.
static __device__ __forceinline__ void atomic_add_f32(float* p, float v) {
  asm volatile("global_atomic_add_f32 %0, %1, off scope:SCOPE_DEV"
               :
               : "v"(p), "v"(v)
               : "memory");
}

// ---------------- setup: degree, d^{-1/2}, edge norm + int32 indices ----------------
__global__ __launch_bounds__(256) void k_degree(const long long* __restrict__ ei,
                                                float* __restrict__ deg) {
  int e = blockIdx.x * 256 + threadIdx.x;
  if (e < NE) {
    int d = (int)ei[NE + e];  // dst
    atomic_add_f32(&deg[d], 1.0f);
  }
}

__global__ __launch_bounds__(256) void k_dinv(float* __restrict__ deg) {
  int n = blockIdx.x * 256 + threadIdx.x;
  if (n < NN) {
    float d = deg[n];
    deg[n] = (d > 0.0f) ? rsqrtf(fmaxf(d, 1.0f)) : 0.0f;
  }
}

__global__ __launch_bounds__(256) void k_edgeprep(const long long* __restrict__ ei,
                                                  const float* __restrict__ dinv,
                                                  int* __restrict__ src, int* __restrict__ dst,
                                                  float* __restrict__ norm) {
  int e = blockIdx.x * 256 + threadIdx.x;
  if (e < NE) {
    int s = (int)ei[e];
    int d = (int)ei[NE + e];
    src[e] = s;
    dst[e] = d;
    norm[e] = dinv[s] * dinv[d];
  }
}

// ---------------- complex GEMM: h = (xr + i·xi)(Wr + i·Wi), fp32 WMMA ----------------
// Block = one of 3125 row-tiles of 16 rows; 8 waves, wave w owns output cols [16w,16w+16).
// Writes h into BOTH the Taylor-carry buffers (tr,ti) and accumulators (ar,ai).
__global__ __launch_bounds__(256) void k_cgemm(const float* __restrict__ xr,
                                               const float* __restrict__ xi,
                                               const float* __restrict__ Wr,
                                               const float* __restrict__ Wi,
                                               float* __restrict__ tr, float* __restrict__ ti,
                                               float* __restrict__ ar, float* __restrict__ ai) {
  __shared__ float lr[16 * DD];
  __shared__ float li[16 * DD];
  const int tid = threadIdx.x;
  const size_t rowBase = (size_t)blockIdx.x * 16 * DD;
#pragma unroll
  for (int j = 0; j < 8; ++j) {
    int idx = tid + j * 256;
    lr[idx] = xr[rowBase + idx];
    li[idx] = xi[rowBase + idx];
  }
  __syncthreads();

  const int wave = tid >> 5;
  const int lane = tid & 31;
  const int m = lane & 15;     // A row / B,C,D col index within tile
  const int half = lane >> 4;  // K-pair select for A/B; M-half select for C/D
  const int col = wave * 16 + m;

  v8f c_rr = {}, c_ii = {}, c_ri = {}, c_ir = {};
#pragma unroll
  for (int k0 = 0; k0 < DD; k0 += 4) {
    const int ka = k0 + half * 2;
    v2f a_r = *(const v2f*)&lr[m * DD + ka];
    v2f a_i = *(const v2f*)&li[m * DD + ka];
    const float* pr = Wr + ka * DD + col;
    const float* pi = Wi + ka * DD + col;
    v2f b_r = {pr[0], pr[DD]};
    v2f b_i = {pi[0], pi[DD]};
    c_rr = __builtin_amdgcn_wmma_f32_16x16x4_f32(false, a_r, false, b_r, (short)0, c_rr, false, false);
    c_ii = __builtin_amdgcn_wmma_f32_16x16x4_f32(false, a_i, false, b_i, (short)0, c_ii, false, false);
    c_ri = __builtin_amdgcn_wmma_f32_16x16x4_f32(false, a_r, false, b_i, (short)0, c_ri, false, false);
    c_ir = __builtin_amdgcn_wmma_f32_16x16x4_f32(false, a_i, false, b_r, (short)0, c_ir, false, false);
  }
#pragma unroll
  for (int v = 0; v < 8; ++v) {
    int mrow = blockIdx.x * 16 + v + 8 * half;
    size_t o = (size_t)mrow * DD + col;
    float hr = c_rr[v] - c_ii[v];
    float hi = c_ri[v] + c_ir[v];
    tr[o] = hr; ti[o] = hi;
    ar[o] = hr; ai[o] = hi;
  }
}

// ---------------- propagate: wave-per-edge gather/scale/scatter-add (r & i fused) ----
__global__ __launch_bounds__(256) void k_scatter(const float4* __restrict__ hr,
                                                 const float4* __restrict__ hi,
                                                 const int* __restrict__ src,
                                                 const int* __restrict__ dst,
                                                 const float* __restrict__ norm,
                                                 float* __restrict__ outr,
                                                 float* __restrict__ outi) {
  int e = blockIdx.x * 8 + (threadIdx.x >> 5);
  if (e >= NE) return;
  // e is wave-uniform by construction; make it provably uniform so that
  // src/dst/norm loads become SMEM scalar loads and address math goes to SALU.
  e = __builtin_amdgcn_readfirstlane(e);
  int lane = threadIdx.x & 31;
  int s = src[e];
  int d = dst[e];
  float w = norm[e];
  float4 vr = hr[(size_t)s * 32 + lane];
  float4 vi = hi[(size_t)s * 32 + lane];
  float* pr = outr + (size_t)d * DD + lane * 4;
  float* pi = outi + (size_t)d * DD + lane * 4;
  atomic_add_f32(pr + 0, vr.x * w);
  atomic_add_f32(pr + 1, vr.y * w);
  atomic_add_f32(pr + 2, vr.z * w);
  atomic_add_f32(pr + 3, vr.w * w);
  atomic_add_f32(pi + 0, vi.x * w);
  atomic_add_f32(pi + 1, vi.y * w);
  atomic_add_f32(pi + 2, vi.z * w);
  atomic_add_f32(pi + 3, vi.w * w);
}

// ---------------- Taylor step update: t = scat/t ; acc += t ; scat = 0 ----------------
__global__ __launch_bounds__(256) void k_update(float4* __restrict__ tr, float4* __restrict__ ti,
                                                float4* __restrict__ sr, float4* __restrict__ si,
                                                float4* __restrict__ ar, float4* __restrict__ ai,
                                                float invt) {
  int i = blockIdx.x * 256 + threadIdx.x;  // NN*DD/4 = 1,600,000 exactly
  float4 vr = sr[i];
  float4 vi = si[i];
  vr.x *= invt; vr.y *= invt; vr.z *= invt; vr.w *= invt;
  vi.x *= invt; vi.y *= invt; vi.z *= invt; vi.w *= invt;
  tr[i] = vr;
  ti[i] = vi;
  float4 cr = ar[i];
  float4 ci = ai[i];
  cr.x += vr.x; cr.y += vr.y; cr.z += vr.z; cr.w += vr.w;
  ci.x += vi.x; ci.y += vi.y; ci.z += vi.z; ci.w += vi.w;
  ar[i] = cr;
  ai[i] = ci;
  float4 z = {0.f, 0.f, 0.f, 0.f};
  sr[i] = z;
  si[i] = z;
}

// ---------------- complex ReLU in place ----------------
__global__ __launch_bounds__(256) void k_relu(float4* __restrict__ a, float4* __restrict__ b) {
  int i = blockIdx.x * 256 + threadIdx.x;
  float4 x = a[i];
  float4 y = b[i];
  x.x = fmaxf(x.x, 0.f); x.y = fmaxf(x.y, 0.f); x.z = fmaxf(x.z, 0.f); x.w = fmaxf(x.w, 0.f);
  y.x = fmaxf(y.x, 0.f); y.y = fmaxf(y.y, 0.f); y.z = fmaxf(y.z, 0.f); y.w = fmaxf(y.w, 0.f);
  a[i] = x;
  b[i] = y;
}

// ---------------- final: out = x @ Wo + bo (fp32 WMMA) ----------------
__global__ __launch_bounds__(256) void k_outgemm(const float* __restrict__ x,
                                                 const float* __restrict__ W,
                                                 const float* __restrict__ bias,
                                                 float* __restrict__ out) {
  __shared__ float lx[16 * DD];
  const int tid = threadIdx.x;
  const size_t rowBase = (size_t)blockIdx.x * 16 * DD;
#pragma unroll
  for (int j = 0; j < 8; ++j) {
    int idx = tid + j * 256;
    lx[idx] = x[rowBase + idx];
  }
  __syncthreads();

  const int wave = tid >> 5;
  const int lane = tid & 31;
  const int m = lane & 15;
  const int half = lane >> 4;
  const int col = wave * 16 + m;

  v8f c = {};
#pragma unroll
  for (int k0 = 0; k0 < DD; k0 += 4) {
    const int ka = k0 + half * 2;
    v2f a = *(const v2f*)&lx[m * DD + ka];
    const float* pw = W + ka * DD + col;
    v2f b = {pw[0], pw[DD]};
    c = __builtin_amdgcn_wmma_f32_16x16x4_f32(false, a, false, b, (short)0, c, false, false);
  }
  float bb = bias[col];
#pragma unroll
  for (int v = 0; v < 8; ++v) {
    int mrow = blockIdx.x * 16 + v + 8 * half;
    out[(size_t)mrow * DD + col] = c[v] + bb;
  }
}

extern "C" void kernel_launch(void* const* d_in, const int* in_sizes, int n_in,
                              void* d_out, int out_size, void* d_ws, size_t ws_size,
                              hipStream_t stream) {
  (void)in_sizes; (void)n_in; (void)out_size; (void)ws_size;
  const float* x_real = (const float*)d_in[0];
  const float* x_imag = (const float*)d_in[1];
  const float* Wr[3] = {(const float*)d_in[2], (const float*)d_in[4], (const float*)d_in[6]};
  const float* Wi[3] = {(const float*)d_in[3], (const float*)d_in[5], (const float*)d_in[7]};
  const float* Wo = (const float*)d_in[8];
  const float* bo = (const float*)d_in[9];
  const long long* ei = (const long long*)d_in[10];
  float* out = (float*)d_out;

  char* ws = (char*)d_ws;
  size_t off = 0;
  auto alloc = [&](size_t bytes) -> void* {
    void* p = ws + off;
    off = (off + bytes + 255) & ~(size_t)255;
    return p;
  };
  const size_t fb = (size_t)NN * DD * sizeof(float);  // 25.6 MB
  float* deg  = (float*)alloc(NN * sizeof(float));
  int*   src  = (int*)alloc(NE * sizeof(int));
  int*   dst  = (int*)alloc(NE * sizeof(int));
  float* norm = (float*)alloc(NE * sizeof(float));
  float* bufAr = (float*)alloc(fb);
  float* bufAi = (float*)alloc(fb);
  float* bufBr = (float*)alloc(fb);
  float* bufBi = (float*)alloc(fb);
  float* Tr = (float*)alloc(fb);
  float* Ti = (float*)alloc(fb);
  float* Sr = (float*)alloc(fb);
  float* Si = (float*)alloc(fb);

  hipMemsetAsync(deg, 0, NN * sizeof(float), stream);
  hipMemsetAsync(Sr, 0, fb, stream);
  hipMemsetAsync(Si, 0, fb, stream);

  k_degree<<<(NE + 255) / 256, 256, 0, stream>>>(ei, deg);
  k_dinv<<<(NN + 255) / 256, 256, 0, stream>>>(deg);
  k_edgeprep<<<(NE + 255) / 256, 256, 0, stream>>>(ei, deg, src, dst, norm);

  const int nRowTiles = NN / 16;          // 3125 (exact)
  const int nVec4 = NN * DD / 4;          // 1,600,000 (exact)
  const int updBlocks = nVec4 / 256;      // 6250 (exact)
  const int scatBlocks = (NE + 7) / 8;    // 100,000

  const float* inr = x_real;
  const float* ini = x_imag;
  float* accR = bufAr; float* accI = bufAi;
  float* altR = bufBr; float* altI = bufBi;

  for (int L = 0; L < 3; ++L) {
    k_cgemm<<<nRowTiles, 256, 0, stream>>>(inr, ini, Wr[L], Wi[L], Tr, Ti, accR, accI);
    for (int t = 1; t <= TT; ++t) {
      k_scatter<<<scatBlocks, 256, 0, stream>>>((const float4*)Tr, (const float4*)Ti,
                                                src, dst, norm, Sr, Si);
      k_update<<<updBlocks, 256, 0, stream>>>((float4*)Tr, (float4*)Ti, (float4*)Sr, (float4*)Si,
                                              (float4*)accR, (float4*)accI, 1.0f / (float)t);
    }
    k_relu<<<updBlocks, 256, 0, stream>>>((float4*)accR, (float4*)accI);
    inr = accR; ini = accI;
    float* tp;
    tp = accR; accR = altR; altR = tp;
    tp = accI; accI = altI; altI = tp;
  }

  k_outgemm<<<nRowTiles, 256, 0, stream>>>(inr, Wo, bo, out);
}